// PatchCoherentSWDLoss_4406636446083
// MI455X (gfx1250) — compile-verified
//
#include <hip/hip_runtime.h>
#include <math.h>

#define NPATCH 62500
#define NPAD   65536
#define OW     250
#define DDIM   147
#define WPITCH 160
#define IMG_ELEMS (3*256*256)

typedef __attribute__((ext_vector_type(2))) float v2f;
typedef __attribute__((ext_vector_type(8))) float v8f;

__device__ __forceinline__ int iclampi(int v, int lo, int hi) {
    return v < lo ? lo : (v > hi ? hi : v);
}

// ---------------------------------------------------------------------------
// 0) init: zero per-batch accumulators, fill sort pad with max sentinels
// ---------------------------------------------------------------------------
__global__ void swd_init_kernel(unsigned long long* sortbuf, unsigned long long* acc) {
    int t = blockIdx.x * blockDim.x + threadIdx.x;
    if (t < 4) acc[t] = 0ULL;
    const int per = NPAD - NPATCH;  // 3036
    if (t < 8 * per) {
        int a = t / per;
        int r = t - a * per;
        sortbuf[(size_t)a * NPAD + NPATCH + r] = ~0ULL;  // sorts to the end
    }
}

// ---------------------------------------------------------------------------
// 1) weights: w = rand / std(rand, ddof=1), per batch (positive scalar scale
//    -> argsort-order invariant, kept for fidelity). Zero-padded to WPITCH.
// ---------------------------------------------------------------------------
__global__ void swd_weights_kernel(const float* __restrict__ rnd, float* __restrict__ w) {
    __shared__ float sh[256];
    int b = blockIdx.x;
    int t = threadIdx.x;
    float v = (t < DDIM) ? rnd[b * DDIM + t] : 0.0f;
    sh[t] = v;
    __syncthreads();
    for (int st = 128; st > 0; st >>= 1) { if (t < st) sh[t] += sh[t + st]; __syncthreads(); }
    float mean = sh[0] / (float)DDIM;
    __syncthreads();
    float dv = (t < DDIM) ? (v - mean) : 0.0f;
    sh[t] = dv * dv;
    __syncthreads();
    for (int st = 128; st > 0; st >>= 1) { if (t < st) sh[t] += sh[t + st]; __syncthreads(); }
    float stdv = sqrtf(sh[0] / (float)(DDIM - 1));
    if (t < DDIM)               w[b * WPITCH + t] = v / stdv;
    else if (t < WPITCH)        w[b * WPITCH + t] = 0.0f;
}

// ---------------------------------------------------------------------------
// 2) projection via V_WMMA_F32_16X16X4_F32 (fp32 in/out preserves the f32
//    ordering the argsort depends on).
//    Wave = 16 consecutive patches (M); K in 37 chunks of 4 over d=147 padded
//    to 148. B carries the weight chunk in column 0 only; lanes 0/16 hold
//    D[:,0] = patch . w and emit packed stable-sort keys.
//    Addressing: incremental (kw,kh,ch) carry chain, no divides, no EXEC
//    divergence in the loop. Pad element k=147 is killed by w[147]==0, so its
//    A load is redirected to the (valid) k=146 address.
// ---------------------------------------------------------------------------
__global__ void swd_proj_kernel(const float* __restrict__ x,
                                const float* __restrict__ y,
                                const float* __restrict__ w,
                                unsigned long long* __restrict__ sortbuf) {
    int a   = blockIdx.y;          // 0..7 : (batch<<1)|img
    int b   = a >> 1;
    const float* img = ((a & 1) ? y : x) + (size_t)b * IMG_ELEMS;
    const float* wb  = w + b * WPITCH;

    int wave = blockIdx.x * (blockDim.x >> 5) + (threadIdx.x >> 5);  // wave-uniform
    int n0 = wave * 16;
    if (n0 >= NPATCH) return;      // uniform per wave -> EXEC stays full

    int lane = threadIdx.x & 31;
    int mloc = lane & 15;
    int half = lane >> 4;          // 0: K 0,1 ; 1: K 2,3 (16x4 f32 A layout)
    bool col0 = (mloc == 0);

    int n  = n0 + mloc; if (n > NPATCH - 1) n = NPATCH - 1;   // clamp, no EXEC change
    int oy = n / OW;
    int ox = n - oy * OW;
    const float* pbase = img + oy * 256 + ox;

    // element-0 walker state: k0 = 4*i + 2*half -> (ch,kh,kw), off in floats
    int kw = 2 * half;
    int kh = 0;
    int k0 = 2 * half;
    int off0 = kw;                 // ch*65536 + kh*256 + kw

    v8f c = {0.f, 0.f, 0.f, 0.f, 0.f, 0.f, 0.f, 0.f};

    #pragma unroll 1
    for (int i = 0; i < 37; ++i) {
        // element 1 = element 0 advanced by one k (carry through kw/kh)
        int k1 = k0 + 1;
        int d1 = (kw == 6) ? ((kh == 6) ? (65536 - 6 * 256 - 6) : (256 - 6)) : 1;
        int off1 = off0 + d1;
        off1 = (k1 > 146) ? off0 : off1;   // k=147: dead via w[147]==0, keep addr valid

        v2f av, bv;
        av.x = pbase[off0];
        av.y = pbase[off1];
        float w0 = wb[k0];                 // unconditional loads, then select
        float w1 = wb[k1];                 // k1 <= 147 < WPITCH
        bv.x = col0 ? w0 : 0.0f;
        bv.y = col0 ? w1 : 0.0f;

        // D = A(16x4 f32) * B(4x16 f32) + C   (fp32 accumulate)
        c = __builtin_amdgcn_wmma_f32_16x16x4_f32(
                false, av, false, bv, (short)0, c, false, false);

        // advance element 0 by +4 in k with carries
        k0 += 4;
        kw += 4; off0 += 4;
        if (kw > 6) { kw -= 7; kh += 1; off0 += 256 - 7; }
        if (kh > 6) { kh -= 7;          off0 += 65536 - 7 * 256; }
    }

    if (col0) {  // lanes 0 and 16 carry D column 0 (M = half*8 + r)
        for (int r = 0; r < 8; ++r) {
            int nn = n0 + half * 8 + r;
            if (nn < NPATCH) {
                unsigned int kb = __float_as_uint(c[r]);
                kb = (kb & 0x80000000u) ? ~kb : (kb | 0x80000000u);  // order-preserving
                sortbuf[(size_t)a * NPAD + nn] =
                    ((unsigned long long)kb << 32) | (unsigned int)nn;  // stable tie-break
            }
        }
    }
}

// ---------------------------------------------------------------------------
// 3) bitonic sort of 8 arrays of 65536 packed u64 (ascending == stable argsort)
// ---------------------------------------------------------------------------
__global__ __launch_bounds__(1024)
void swd_sort_local_kernel(unsigned long long* __restrict__ buf) {
    __shared__ unsigned long long s[2048];
    int arr = blockIdx.y, chunk = blockIdx.x;
    unsigned long long* g = buf + (size_t)arr * NPAD + (size_t)chunk * 2048;
    int t = threadIdx.x;
    s[t] = g[t];
    s[t + 1024] = g[t + 1024];
    __syncthreads();
    int base = chunk * 2048;
    for (int k = 2; k <= 2048; k <<= 1) {
        for (int j = k >> 1; j > 0; j >>= 1) {
            int i = ((t & ~(j - 1)) << 1) | (t & (j - 1));
            int l = i | j;
            bool dir = (((base + i) & k) == 0);
            unsigned long long A = s[i], B = s[l];
            bool sw = dir ? (A > B) : (A < B);
            if (sw) { s[i] = B; s[l] = A; }
            __syncthreads();
        }
    }
    g[t] = s[t];
    g[t + 1024] = s[t + 1024];
}

__global__ void swd_sort_gstep_kernel(unsigned long long* __restrict__ buf, int k, int j) {
    int arr = blockIdx.y;
    int t = blockIdx.x * blockDim.x + threadIdx.x;   // < 32768
    int i = ((t & ~(j - 1)) << 1) | (t & (j - 1));
    int l = i | j;
    bool dir = ((i & k) == 0);
    unsigned long long* g = buf + (size_t)arr * NPAD;
    unsigned long long A = g[i], B = g[l];
    bool sw = dir ? (A > B) : (A < B);
    if (sw) { g[i] = B; g[l] = A; }
}

__global__ __launch_bounds__(1024)
void swd_sort_lmerge_kernel(unsigned long long* __restrict__ buf, int k) {
    __shared__ unsigned long long s[2048];
    int arr = blockIdx.y, chunk = blockIdx.x;
    unsigned long long* g = buf + (size_t)arr * NPAD + (size_t)chunk * 2048;
    int t = threadIdx.x;
    s[t] = g[t];
    s[t + 1024] = g[t + 1024];
    __syncthreads();
    bool dir = (((chunk * 2048) & k) == 0);          // k >= 4096 -> uniform per chunk
    for (int j = 1024; j > 0; j >>= 1) {
        int i = ((t & ~(j - 1)) << 1) | (t & (j - 1));
        int l = i | j;
        unsigned long long A = s[i], B = s[l];
        bool sw = dir ? (A > B) : (A < B);
        if (sw) { s[i] = B; s[l] = A; }
        __syncthreads();
    }
    g[t] = s[t];
    g[t + 1024] = s[t + 1024];
}

// ---------------------------------------------------------------------------
// 4) exact replica of the fixed-17-iteration bisect_left on the UNSORTED at[],
//    nearest selection, int64 squared-diff accumulation per batch.
//    sorted idx arrays are the low 32 bits of the packed u64s.
// ---------------------------------------------------------------------------
__global__ void swd_loss_kernel(const unsigned long long* __restrict__ sortbuf,
                                unsigned long long* __restrict__ acc) {
    int b = blockIdx.y;
    int m = blockIdx.x * blockDim.x + threadIdx.x;
    const unsigned int* sx = (const unsigned int*)(sortbuf + (size_t)(2 * b) * NPAD);
    const unsigned int* sy = (const unsigned int*)(sortbuf + (size_t)(2 * b + 1) * NPAD);
    unsigned long long val = 0ULL;
    if (m < NPATCH) {
        int v = (int)sx[2 * m];                     // ai[b][m] (low word, LE)
        int lo = 0, hi = NPATCH;
        for (int it = 0; it < 17; ++it) {           // ceil(log2(62500)) + 1
            int mid = (lo + hi) >> 1;
            int mc = iclampi(mid, 0, NPATCH - 1);
            int am = (int)sy[2 * mc];
            bool go = lo < hi;
            bool pred = am < v;
            if (go && pred)  lo = mid + 1;
            if (go && !pred) hi = mid;
        }
        int idx = lo;
        int ip = iclampi(idx - 1, 0, NPATCH - 1);
        int ic = iclampi(idx,     0, NPATCH - 1);
        int a_prev = (int)sy[2 * ip];
        int a_at   = (int)sy[2 * ic];
        int dp = v - a_prev; if (dp < 0) dp = -dp;
        int da = v - a_at;   if (da < 0) da = -da;
        bool take_prev = (idx > 0) && ((idx == NPATCH) || (dp < da));
        int nearest = take_prev ? a_prev : a_at;
        long long d = (long long)(v - nearest);
        val = (unsigned long long)(d * d);
    }
    __shared__ unsigned long long s[256];
    s[threadIdx.x] = val;
    __syncthreads();
    for (int st = 128; st > 0; st >>= 1) {
        if (threadIdx.x < st) s[threadIdx.x] += s[threadIdx.x + st];
        __syncthreads();
    }
    if (threadIdx.x == 0) atomicAdd(&acc[b], s[0]);
}

// ---------------------------------------------------------------------------
// 5) finalize: mean over batch of (sum / N)
// ---------------------------------------------------------------------------
__global__ void swd_final_kernel(const unsigned long long* __restrict__ acc,
                                 float* __restrict__ out) {
    if (threadIdx.x == 0 && blockIdx.x == 0) {
        double s = 0.0;
        for (int b = 0; b < 4; ++b) s += (double)acc[b] / (double)NPATCH;
        out[0] = (float)(s / 4.0);
    }
}

// ---------------------------------------------------------------------------
extern "C" void kernel_launch(void* const* d_in, const int* in_sizes, int n_in,
                              void* d_out, int out_size, void* d_ws, size_t ws_size,
                              hipStream_t stream) {
    const float* x   = (const float*)d_in[0];   // [4,3,256,256]
    const float* y   = (const float*)d_in[1];   // [4,3,256,256]
    const float* rnd = (const float*)d_in[2];   // [4,147,1]
    float* out = (float*)d_out;

    char* ws = (char*)d_ws;
    unsigned long long* sortbuf = (unsigned long long*)ws;                 // 8*65536*8 = 4 MiB
    float*              w       = (float*)(ws + (size_t)8 * NPAD * 8);    // 4*160*4
    unsigned long long* acc     = (unsigned long long*)(ws + (size_t)8 * NPAD * 8 + 4096);

    // 0) init pad + accumulators
    swd_init_kernel<<<dim3(96), dim3(256), 0, stream>>>(sortbuf, acc);

    // 1) normalized projection weights
    swd_weights_kernel<<<dim3(4), dim3(256), 0, stream>>>(rnd, w);

    // 2) WMMA projection -> packed sort keys
    {
        int waves = (NPATCH + 15) / 16;               // 3907
        int blocks = (waves + 7) / 8;                 // 8 waves / 256-thread block
        swd_proj_kernel<<<dim3(blocks, 8), dim3(256), 0, stream>>>(x, y, w, sortbuf);
    }

    // 3) bitonic sort: 8 arrays of 65536 u64
    swd_sort_local_kernel<<<dim3(32, 8), dim3(1024), 0, stream>>>(sortbuf);
    for (int k = 4096; k <= NPAD; k <<= 1) {
        for (int j = k >> 1; j >= 2048; j >>= 1)
            swd_sort_gstep_kernel<<<dim3(128, 8), dim3(256), 0, stream>>>(sortbuf, k, j);
        swd_sort_lmerge_kernel<<<dim3(32, 8), dim3(1024), 0, stream>>>(sortbuf, k);
    }

    // 4) bisect + nearest + int64 accumulation
    {
        int blocks = (NPATCH + 255) / 256;            // 245
        swd_loss_kernel<<<dim3(blocks, 4), dim3(256), 0, stream>>>(sortbuf, acc);
    }

    // 5) finalize
    swd_final_kernel<<<dim3(1), dim3(64), 0, stream>>>(acc, out);
}